// GIG_Transformer_46102178955386
// MI455X (gfx1250) — compile-verified
//
#include <hip/hip_runtime.h>
#include <hip/hip_bf16.h>

// Problem constants (from reference)
#define NN 50000
#define EE 800000
#define INF_DIM 128
#define HH 8
#define CC 16
#define EDIM 16
#define HC 128   // H*C

typedef __attribute__((ext_vector_type(2))) float v2f;
typedef __attribute__((ext_vector_type(8))) float v8f;

// ---------------------------------------------------------------------------
// Kernel 1: fused node GEMMs  out = x @ W + b  for W in {w_q, w_k, w_v, w_skip}
// fp32-exact via V_WMMA_F32_16X16X4_F32 (16x16 tile, K stepped by 4).
// One wave owns a 16-row tile and loops over the 8 column tiles.
// blockDim = 256 (8 waves); gridDim.x covers row tiles; gridDim.y = matrix id.
// ---------------------------------------------------------------------------
__global__ void node_gemm_wmma(const float* __restrict__ x,
                               const float* __restrict__ w_q, const float* __restrict__ b_q,
                               const float* __restrict__ w_k, const float* __restrict__ b_k,
                               const float* __restrict__ w_v, const float* __restrict__ b_v,
                               const float* __restrict__ w_s, const float* __restrict__ b_s,
                               float* __restrict__ qo, float* __restrict__ ko,
                               float* __restrict__ vo, float* __restrict__ so)
{
    const int wave = threadIdx.x >> 5;
    const int lane = threadIdx.x & 31;
    const int tile = blockIdx.x * 8 + wave;          // 16-row tile index
    const int n_tiles = NN / 16;                     // 3125 exactly
    if (tile >= n_tiles) return;                     // wave-uniform: EXEC stays all-1s

    const float* W; const float* B; float* O;
    switch (blockIdx.y) {
        case 0:  W = w_q; B = b_q; O = qo; break;
        case 1:  W = w_k; B = b_k; O = ko; break;
        case 2:  W = w_v; B = b_v; O = vo; break;
        default: W = w_s; B = b_s; O = so; break;
    }

    const int r0    = tile * 16;
    const int arow  = r0 + (lane & 15);
    const int kbase = (lane < 16) ? 0 : 2;           // K-half split per ISA A-layout

    // Preload all A fragments for the full K=128 (32 steps of 4) — 64 VGPRs.
    v2f afrag[32];
#pragma unroll
    for (int ks = 0; ks < 32; ++ks) {
        const float* p = x + (size_t)arow * INF_DIM + ks * 4 + kbase;
        afrag[ks] = *(const v2f*)p;                  // 8B-aligned (kbase even)
    }

    const int ncl = lane & 15;
#pragma unroll 1
    for (int nt = 0; nt < 8; ++nt) {
        const int n = nt * 16 + ncl;
        v8f acc = {};
#pragma unroll
        for (int ks = 0; ks < 32; ++ks) {
            v2f b;
            b.x = W[(ks * 4 + kbase + 0) * HC + n];
            b.y = W[(ks * 4 + kbase + 1) * HC + n];
            acc = __builtin_amdgcn_wmma_f32_16x16x4_f32(
                false, afrag[ks], false, b, (short)0, acc, false, false);
        }
        const float bias = B[n];
        const int rbase = r0 + ((lane < 16) ? 0 : 8);
#pragma unroll
        for (int i = 0; i < 8; ++i)
            O[(size_t)(rbase + i) * HC + n] = acc[i] + bias;
    }
}

// ---------------------------------------------------------------------------
// Init: segmax = -inf, segsum = 0   (N*H entries each)
// ---------------------------------------------------------------------------
__global__ void seg_init(float* __restrict__ segmax, float* __restrict__ segsum)
{
    int i = blockIdx.x * 256 + threadIdx.x;
    if (i >= NN * HH) return;
    segmax[i] = __int_as_float(0xFF800000);          // -inf
    segsum[i] = 0.0f;
}

// Sign-split float atomic-max (works with -inf init; standard monotone trick).
__device__ __forceinline__ void atomicMaxFloat(float* addr, float val)
{
    if (val >= 0.0f) atomicMax((int*)addr, __float_as_int(val));
    else             atomicMin((unsigned int*)addr, __float_as_uint(val));
}

// Compute e[h, 0..15] for one edge from LDS-cached w_e (8KB).
__device__ __forceinline__ void edge_bias(const float* __restrict__ edge_attr, int e, int h,
                                          const float* __restrict__ we_s, float eb[CC])
{
    float ea[EDIM] __attribute__((aligned(16)));
#pragma unroll
    for (int i = 0; i < 4; ++i)
        reinterpret_cast<float4*>(ea)[i] =
            reinterpret_cast<const float4*>(edge_attr + (size_t)e * EDIM)[i];

#pragma unroll
    for (int c = 0; c < CC; ++c) eb[c] = 0.0f;
#pragma unroll
    for (int d = 0; d < EDIM; ++d) {
        const float ad = ea[d];
        float wv[CC] __attribute__((aligned(16)));
        const float4* wr = reinterpret_cast<const float4*>(we_s + d * HC + h * CC);
#pragma unroll
        for (int i = 0; i < 4; ++i) reinterpret_cast<float4*>(wv)[i] = wr[i];
#pragma unroll
        for (int c = 0; c < CC; ++c) eb[c] = fmaf(ad, wv[c], eb[c]);
    }
}

// ---------------------------------------------------------------------------
// Pass 1: per (edge, head) logit  alpha = q[dst]·(k[src]+e) / 4 ; seg-max
// ---------------------------------------------------------------------------
__global__ void edge_alpha_kernel(const int* __restrict__ ei,
                                  const float* __restrict__ edge_attr,
                                  const float* __restrict__ w_e,
                                  const float* __restrict__ q,
                                  const float* __restrict__ k,
                                  float* __restrict__ alpha,
                                  float* __restrict__ segmax)
{
    __shared__ float we_s[EDIM * HC] __attribute__((aligned(16)));
    for (int i = threadIdx.x; i < EDIM * HC; i += 256) we_s[i] = w_e[i];
    __syncthreads();

    const int idx = blockIdx.x * 256 + threadIdx.x;
    if (idx >= EE * HH) return;
    const int e = idx >> 3, h = idx & 7;
    const int src = ei[e];
    const int dst = ei[EE + e];

    float eb[CC];
    edge_bias(edge_attr, e, h, we_s, eb);

    const float* qp = q + (size_t)dst * HC + h * CC;
    const float* kp = k + (size_t)src * HC + h * CC;
    float acc = 0.0f;
#pragma unroll
    for (int c = 0; c < CC; ++c)
        acc = fmaf(qp[c], kp[c] + eb[c], acc);
    acc *= 0.25f;                                    // 1/sqrt(C)

    alpha[idx] = acc;
    atomicMaxFloat(&segmax[dst * HH + h], acc);
}

// ---------------------------------------------------------------------------
// Pass 2: alpha = exp(alpha - segmax[dst]); seg-sum
// ---------------------------------------------------------------------------
__global__ void edge_softmax_kernel(const int* __restrict__ ei,
                                    float* __restrict__ alpha,
                                    const float* __restrict__ segmax,
                                    float* __restrict__ segsum)
{
    const int idx = blockIdx.x * 256 + threadIdx.x;
    if (idx >= EE * HH) return;
    const int e = idx >> 3, h = idx & 7;
    const int dst = ei[EE + e];
    const float a = expf(alpha[idx] - segmax[dst * HH + h]);
    alpha[idx] = a;
    atomicAdd(&segsum[dst * HH + h], a);
}

// ---------------------------------------------------------------------------
// Pass 3: out[dst] += (v[src] + e) * alpha / (segsum[dst] + 1e-16)
// (out was pre-initialized with the skip GEMM result.)
// ---------------------------------------------------------------------------
__global__ void edge_message_kernel(const int* __restrict__ ei,
                                    const float* __restrict__ edge_attr,
                                    const float* __restrict__ w_e,
                                    const float* __restrict__ v,
                                    const float* __restrict__ alpha,
                                    const float* __restrict__ segsum,
                                    float* __restrict__ out)
{
    __shared__ float we_s[EDIM * HC] __attribute__((aligned(16)));
    for (int i = threadIdx.x; i < EDIM * HC; i += 256) we_s[i] = w_e[i];
    __syncthreads();

    const int idx = blockIdx.x * 256 + threadIdx.x;
    if (idx >= EE * HH) return;
    const int e = idx >> 3, h = idx & 7;
    const int src = ei[e];
    const int dst = ei[EE + e];

    const float wgt = alpha[idx] / (segsum[dst * HH + h] + 1e-16f);

    float eb[CC];
    edge_bias(edge_attr, e, h, we_s, eb);

    const float* vp = v + (size_t)src * HC + h * CC;
    float* op = out + (size_t)dst * HC + h * CC;
#pragma unroll
    for (int c = 0; c < CC; ++c)
        atomicAdd(&op[c], (vp[c] + eb[c]) * wgt);
}

// ---------------------------------------------------------------------------
extern "C" void kernel_launch(void* const* d_in, const int* in_sizes, int n_in,
                              void* d_out, int out_size, void* d_ws, size_t ws_size,
                              hipStream_t stream)
{
    const float* x         = (const float*)d_in[0];
    const int*   ei        = (const int*)  d_in[1];
    const float* edge_attr = (const float*)d_in[2];
    const float* w_q       = (const float*)d_in[3];
    const float* b_q       = (const float*)d_in[4];
    const float* w_k       = (const float*)d_in[5];
    const float* b_k       = (const float*)d_in[6];
    const float* w_v       = (const float*)d_in[7];
    const float* b_v       = (const float*)d_in[8];
    const float* w_e       = (const float*)d_in[9];
    const float* w_skip    = (const float*)d_in[10];
    const float* b_skip    = (const float*)d_in[11];
    float* out = (float*)d_out;

    // Workspace carve-out (~106 MB)
    float* q      = (float*)d_ws;                    // N*128
    float* k      = q + (size_t)NN * HC;             // N*128
    float* v      = k + (size_t)NN * HC;             // N*128
    float* alpha  = v + (size_t)NN * HC;             // E*H
    float* segmax = alpha + (size_t)EE * HH;         // N*H
    float* segsum = segmax + (size_t)NN * HH;        // N*H

    // 1) q,k,v,skip GEMMs (skip writes straight into d_out)
    dim3 ggrid((NN / 16 + 7) / 8, 4, 1);
    node_gemm_wmma<<<ggrid, 256, 0, stream>>>(x, w_q, b_q, w_k, b_k, w_v, b_v,
                                              w_skip, b_skip, q, k, v, out);

    // 2) init softmax accumulators
    seg_init<<<(NN * HH + 255) / 256, 256, 0, stream>>>(segmax, segsum);

    // 3) logits + segment max
    const int eh_blocks = (EE * HH + 255) / 256;     // 25000
    edge_alpha_kernel<<<eh_blocks, 256, 0, stream>>>(ei, edge_attr, w_e, q, k,
                                                     alpha, segmax);

    // 4) exp + segment sum
    edge_softmax_kernel<<<eh_blocks, 256, 0, stream>>>(ei, alpha, segmax, segsum);

    // 5) normalized message scatter-add into out
    edge_message_kernel<<<eh_blocks, 256, 0, stream>>>(ei, edge_attr, w_e, v,
                                                       alpha, segsum, out);
}